// InverseCompositional_22462678958151
// MI455X (gfx1250) — compile-verified
//
#include <hip/hip_runtime.h>
#include <stdint.h>

// ---------------- problem constants ----------------
#define IMW   1024
#define IMH   1024
#define NB    8
#define NDELTA 10
#define TOLV  1e-3f
#define MAXIT 30
#define TILE_ROWS 8
#define ROW_BASE  8          // tiles cover rows [8,1016); valid rows [10,1014)
#define NTILES 126           // 126*8 = 1008 rows
#define STRIP_ROWS 10        // rows staged per tile (TILE_ROWS + 2)

// ---------------- workspace layout (floats) ----------------
#define WS_BPART 0           // 16 rows x 128 (126 tiles + 2 zero pad)
#define WS_HPART 2048        // 8*126*3
#define WS_P     5120        // 16
#define WS_PSNAP 5136        // 16
#define WS_HINV  5152        // 8*4
#define WS_ERR   5184
#define WS_ACT   5185

typedef uint32_t u32x4 __attribute__((ext_vector_type(4)));
typedef int      i32x8 __attribute__((ext_vector_type(8)));
typedef int      i32x4 __attribute__((ext_vector_type(4)));
typedef float    v2f   __attribute__((ext_vector_type(2)));
typedef float    v8f   __attribute__((ext_vector_type(8)));

// ---- Tensor Data Mover: 2D fp32 tile, global -> LDS (D# per CDNA5 ISA) ----
__device__ __forceinline__ void tdm_load_tile_f32(const float* gsrc, uint32_t lds_off,
                                                  int tile_w, int tile_h,
                                                  int tensor_w, int tensor_h,
                                                  long long row_stride_elems) {
  unsigned long long ga = (unsigned long long)(uintptr_t)gsrc;
  u32x4 g0;
  g0[0] = 1u;                                               // count=1, user mode
  g0[1] = (uint32_t)__builtin_amdgcn_readfirstlane((int)lds_off);
  g0[2] = (uint32_t)__builtin_amdgcn_readfirstlane((int)(uint32_t)(ga & 0xFFFFFFFFull));
  g0[3] = (uint32_t)__builtin_amdgcn_readfirstlane(
            (int)((uint32_t)((ga >> 32) & 0x01FFFFFFull) | (2u << 30)));  // addr[56:32] | type=2
  i32x8 g1;
  g1[0] = (2 << 16);                                        // data_size = 4 bytes
  g1[1] = (int)((tensor_w & 0xFFFF) << 16);                 // tensor_dim0[15:0]
  g1[2] = (int)(((unsigned)tensor_w >> 16) & 0xFFFF) | (int)((tensor_h & 0xFFFF) << 16);
  g1[3] = (int)(((unsigned)tensor_h >> 16) & 0xFFFF) | (int)((tile_w & 0xFFFF) << 16);
  g1[4] = (tile_h & 0xFFFF);                                // tile_dim1; tile_dim2=0 (2D)
  g1[5] = (int)(row_stride_elems & 0xFFFFFFFFll);           // tensor_dim0_stride[31:0]
  g1[6] = (int)((row_stride_elems >> 32) & 0xFFFFll);       // stride[47:32]; dim1_stride lo = 0
  g1[7] = 0;
  i32x4 g2 = {0, 0, 0, 0};                                  // 2D: dims 2/3 unused
  i32x4 g3 = {0, 0, 0, 0};
  i32x8 g4 = {0, 0, 0, 0, 0, 0, 0, 0};                      // extended words (unused here)
  __builtin_amdgcn_tensor_load_to_lds(g0, g1, g2, g3, g4, 0);
}

__device__ __forceinline__ int clampi(int v, int lo, int hi) {
  return v < lo ? lo : (v > hi ? hi : v);
}

// ---------------- Hessian partials (once) ----------------
__global__ __launch_bounds__(256) void kHess(const float* __restrict__ I1, float* __restrict__ ws) {
  __shared__ float h0[256], h1[256], h2[256];
  const int b  = blockIdx.y;
  const int y0 = ROW_BASE + blockIdx.x * TILE_ROWS;
  const float* I1b = I1 + ((size_t)b << 20);
  float a = 0.f, c = 0.f, d = 0.f;
  for (int r = 0; r < TILE_ROWS; ++r) {
    const int y = y0 + r;
    if (y < NDELTA || y >= IMH - NDELTA) continue;
    const float* rm = I1b + (size_t)(y - 1) * IMW;
    const float* rc = I1b + (size_t)y * IMW;
    const float* rp = I1b + (size_t)(y + 1) * IMW;
    for (int x = NDELTA + threadIdx.x; x < IMW - NDELTA; x += 256) {
      const float gx = (rc[x + 1] - rc[x - 1]) * 0.5f;
      const float gy = (rp[x] - rm[x]) * 0.5f;
      a = fmaf(gx, gx, a); c = fmaf(gx, gy, c); d = fmaf(gy, gy, d);
    }
  }
  h0[threadIdx.x] = a; h1[threadIdx.x] = c; h2[threadIdx.x] = d;
  __syncthreads();
  for (int off = 128; off > 0; off >>= 1) {
    if ((int)threadIdx.x < off) {
      h0[threadIdx.x] += h0[threadIdx.x + off];
      h1[threadIdx.x] += h1[threadIdx.x + off];
      h2[threadIdx.x] += h2[threadIdx.x + off];
    }
    __syncthreads();
  }
  if (threadIdx.x == 0) {
    const int base = WS_HPART + (b * NTILES + blockIdx.x) * 3;
    ws[base + 0] = h0[0]; ws[base + 1] = h1[0]; ws[base + 2] = h2[0];
  }
}

// ---------------- init: 2x2 inverse, state, zero partial pads ----------------
__global__ __launch_bounds__(256) void kInit(const float* __restrict__ pin, float* __restrict__ ws) {
  const int t = threadIdx.x;
  for (int i = t; i < 16 * 128; i += 256) ws[WS_BPART + i] = 0.f;
  if (t < 16) { ws[WS_P + t] = pin[t]; ws[WS_PSNAP + t] = pin[t]; }
  if (t == 0) { ws[WS_ERR] = 1e10f; ws[WS_ACT] = 1.f; }
  if (t < NB) {
    float a = 0.f, c = 0.f, d = 0.f;
    for (int i = 0; i < NTILES; ++i) {
      const int base = WS_HPART + (t * NTILES + i) * 3;
      a += ws[base + 0]; c += ws[base + 1]; d += ws[base + 2];
    }
    const float inv = 1.f / (a * d - c * c);
    ws[WS_HINV + t * 4 + 0] =  d * inv;
    ws[WS_HINV + t * 4 + 1] = -c * inv;
    ws[WS_HINV + t * 4 + 2] = -c * inv;
    ws[WS_HINV + t * 4 + 3] =  a * inv;
  }
}

// ---------------- hot loop: TDM-staged warp + bvec partials ----------------
__global__ __launch_bounds__(256) void kIter(const float* __restrict__ I1,
                                             const float* __restrict__ I2,
                                             float* __restrict__ ws) {
  if (ws[WS_ACT] == 0.f) return;   // uniform, deterministic gate
  __shared__ float sI1[STRIP_ROWS * IMW];
  __shared__ float sI2[STRIP_ROWS * IMW];
  __shared__ float r0[256], r1[256];
  const int b  = blockIdx.y;
  const int y0 = ROW_BASE + blockIdx.x * TILE_ROWS;
  const float tx = ws[WS_P + 2 * b + 0];
  const float ty = ws[WS_P + 2 * b + 1];
  const int fty = (int)floorf(ty);
  const int ys  = clampi(y0 + fty, 0, IMH - STRIP_ROWS);   // staged I2 rows [ys, ys+9]
  const float* I1b = I1 + ((size_t)b << 20);
  const float* I2b = I2 + ((size_t)b << 20);

  if (threadIdx.x == 0) {   // wave 0 issues both DMAs (TDM ignores EXEC)
    tdm_load_tile_f32(I1b + (size_t)(y0 - 1) * IMW, (uint32_t)(uintptr_t)(void*)sI1,
                      IMW, STRIP_ROWS, IMW, IMH, IMW);
    tdm_load_tile_f32(I2b + (size_t)ys * IMW,       (uint32_t)(uintptr_t)(void*)sI2,
                      IMW, STRIP_ROWS, IMW, IMH, IMW);
    __builtin_amdgcn_s_wait_tensorcnt(0);
  }
  __syncthreads();

  float s0 = 0.f, s1 = 0.f;
  for (int r = 0; r < TILE_ROWS; ++r) {
    const int y = y0 + r;
    if (y < NDELTA || y >= IMH - NDELTA) continue;   // zero-gradient border: no contribution
    const float yw  = (float)y + ty;
    const float y0f = floorf(yw);
    const float fy  = yw - y0f;
    const int yi0 = clampi((int)y0f, 0, IMH - 1);
    const int yi1 = yi0 + 1 > IMH - 1 ? IMH - 1 : yi0 + 1;
    const bool vy = (yw >= 0.f) && (yw <= (float)(IMH - 1));
    const float* s2a = sI2 + (yi0 - ys) * IMW;
    const float* s2b = sI2 + (yi1 - ys) * IMW;
    const float* i1m = sI1 + r * IMW;          // row y-1
    const float* i1c = sI1 + (r + 1) * IMW;    // row y
    const float* i1p = sI1 + (r + 2) * IMW;    // row y+1
    for (int x = NDELTA + threadIdx.x; x < IMW - NDELTA; x += 256) {
      const float xw  = (float)x + tx;
      const float x0f = floorf(xw);
      const float fx  = xw - x0f;
      const int xi0 = clampi((int)x0f, 0, IMW - 1);
      const int xi1 = xi0 + 1 > IMW - 1 ? IMW - 1 : xi0 + 1;
      if (!(vy && xw >= 0.f && xw <= (float)(IMW - 1))) continue;
      const float v00 = s2a[xi0], v01 = s2a[xi1];
      const float v10 = s2b[xi0], v11 = s2b[xi1];
      const float iw = v00 * (1.f - fx) * (1.f - fy) + v01 * fx * (1.f - fy)
                     + v10 * (1.f - fx) * fy         + v11 * fx * fy;
      const float di = iw - i1c[x];
      const float gx = (i1c[x + 1] - i1c[x - 1]) * 0.5f;
      const float gy = (i1p[x] - i1m[x]) * 0.5f;
      s0 = fmaf(gx, di, s0);
      s1 = fmaf(gy, di, s1);
    }
  }
  r0[threadIdx.x] = s0; r1[threadIdx.x] = s1;
  __syncthreads();
  for (int off = 128; off > 0; off >>= 1) {
    if ((int)threadIdx.x < off) {
      r0[threadIdx.x] += r0[threadIdx.x + off];
      r1[threadIdx.x] += r1[threadIdx.x + off];
    }
    __syncthreads();
  }
  if (threadIdx.x == 0) {
    ws[WS_BPART + (2 * b + 0) * 128 + blockIdx.x] = r0[0];
    ws[WS_BPART + (2 * b + 1) * 128 + blockIdx.x] = r1[0];
  }
}

// ------- per-iteration solve: WMMA f32 16x16x4 reduction of tile partials -------
__global__ __launch_bounds__(32) void kUpdate(float* __restrict__ ws) {
  if (ws[WS_ACT] == 0.f) return;
  __shared__ float bsum[16];
  const int lane = threadIdx.x;
  const int row  = lane & 15;
  const float* bp = ws + WS_BPART + row * 128;
  v8f c = {0.f, 0.f, 0.f, 0.f, 0.f, 0.f, 0.f, 0.f};
  v2f ones; ones[0] = 1.f; ones[1] = 1.f;
  const int kb = (lane < 16) ? 0 : 2;               // A 16x4 f32 lane layout
  for (int k = 0; k < 128; k += 4) {                // D = A x ones + C : row sums
    v2f a; a[0] = bp[k + kb]; a[1] = bp[k + kb + 1];
    c = __builtin_amdgcn_wmma_f32_16x16x4_f32(false, a, false, ones, (short)0, c, false, false);
  }
  if (lane == 0)  {                                  // (M=0..7,  N=0) lives in lane 0
#pragma unroll
    for (int i = 0; i < 8; ++i) bsum[i] = c[i];
  }
  if (lane == 16) {                                  // (M=8..15, N=0) lives in lane 16
#pragma unroll
    for (int i = 0; i < 8; ++i) bsum[8 + i] = c[i];
  }
  __syncthreads();
  if (lane == 0) {
    float err2 = 0.f, dpv[16];
    for (int b = 0; b < NB; ++b) {
      const float bx = bsum[2 * b], by = bsum[2 * b + 1];
      const float* h = ws + WS_HINV + 4 * b;
      const float dx = h[0] * bx + h[1] * by;
      const float dy = h[2] * bx + h[3] * by;
      dpv[2 * b] = dx; dpv[2 * b + 1] = dy;
      err2 += dx * dx + dy * dy;
    }
    const float err = sqrtf(err2);
    for (int i = 0; i < 16; ++i) {
      const float pe = ws[WS_P + i];
      ws[WS_PSNAP + i] = pe;          // p at entry of this (active) iteration
      ws[WS_P + i] = pe - dpv[i];     // inverse-compositional translation update
    }
    ws[WS_ERR] = err;
    ws[WS_ACT] = (err > TOLV) ? 1.f : 0.f;
  }
}

// ---------------- final pass: DI/Iw with reference NaN semantics + p/err ----------------
__global__ __launch_bounds__(256) void kFinal(const float* __restrict__ I1,
                                              const float* __restrict__ I2,
                                              const float* __restrict__ ws,
                                              float* __restrict__ dout) {
  const int b = blockIdx.z, y = blockIdx.y;
  const int x = blockIdx.x * 256 + threadIdx.x;
  const float tx = ws[WS_PSNAP + 2 * b + 0];
  const float ty = ws[WS_PSNAP + 2 * b + 1];
  const float* I2b = I2 + ((size_t)b << 20);
  const float xw = (float)x + tx;
  const float yw = (float)y + ty;
  const float x0f = floorf(xw), y0f = floorf(yw);
  const float fx = xw - x0f,    fy = yw - y0f;
  const int xi0 = clampi((int)x0f, 0, IMW - 1);
  const int xi1 = xi0 + 1 > IMW - 1 ? IMW - 1 : xi0 + 1;
  const int yi0 = clampi((int)y0f, 0, IMH - 1);
  const int yi1 = yi0 + 1 > IMH - 1 ? IMH - 1 : yi0 + 1;
  const bool v = (xw >= 0.f) && (xw <= (float)(IMW - 1)) &&
                 (yw >= 0.f) && (yw <= (float)(IMH - 1));
  float iw;
  if (v) {
    const float v00 = I2b[(size_t)yi0 * IMW + xi0], v01 = I2b[(size_t)yi0 * IMW + xi1];
    const float v10 = I2b[(size_t)yi1 * IMW + xi0], v11 = I2b[(size_t)yi1 * IMW + xi1];
    iw = v00 * (1.f - fx) * (1.f - fy) + v01 * fx * (1.f - fy)
       + v10 * (1.f - fx) * fy         + v11 * fx * fy;
  } else {
    iw = __uint_as_float(0x7FC00000u);   // NaN outside image (reference semantics)
  }
  const size_t idx = ((size_t)b << 20) + (size_t)y * IMW + x;
  const float di = iw - I1[idx];         // NaN propagates
  dout[17 + idx] = di;
  dout[17 + (size_t)NB * IMH * IMW + idx] = iw;
  if (blockIdx.x == 0 && blockIdx.y == 0 && blockIdx.z == 0) {
    if (threadIdx.x < 16) dout[threadIdx.x] = ws[WS_P + threadIdx.x];
    if (threadIdx.x == 16) dout[16] = ws[WS_ERR];
  }
}

// ---------------- launch ----------------
extern "C" void kernel_launch(void* const* d_in, const int* in_sizes, int n_in,
                              void* d_out, int out_size, void* d_ws, size_t ws_size,
                              hipStream_t stream) {
  const float* I1 = (const float*)d_in[0];
  const float* I2 = (const float*)d_in[1];
  const float* p  = (const float*)d_in[2];
  // d_in[3] = transform_type (0 = translation), unused
  float* ws  = (float*)d_ws;
  float* out = (float*)d_out;

  kHess<<<dim3(NTILES, NB), 256, 0, stream>>>(I1, ws);
  kInit<<<1, 256, 0, stream>>>(p, ws);
  for (int it = 0; it < MAXIT; ++it) {
    kIter<<<dim3(NTILES, NB), 256, 0, stream>>>(I1, I2, ws);
    kUpdate<<<1, 32, 0, stream>>>(ws);
  }
  kFinal<<<dim3(IMW / 256, IMH, NB), 256, 0, stream>>>(I1, I2, ws, out);
}